// ProbMaxPool_87917980549283
// MI455X (gfx1250) — compile-verified
//
#include <hip/hip_runtime.h>
#include <hip/hip_bf16.h>
#include <stdint.h>

// ProbMaxPool fused kernel for MI455X (gfx1250, wave32).
// Shapes: H[16,64,128,128] f32, MP=2 -> nr=nc=64, K=4.
// out = probs (16M floats) then states (16M floats).

typedef __attribute__((ext_vector_type(2))) float v2f;
typedef __attribute__((ext_vector_type(8))) float v8f;
typedef __attribute__((ext_vector_type(4))) int   v4i;

#if __has_builtin(__builtin_amdgcn_global_load_async_to_lds_b128)
#define USE_ASYNC_LDS 1
#endif

// Address-space casts via integer round-trip (always compiles; on AMDGPU a
// generic global address is value-identical to the AS1 address, and the low
// 32 bits of a generic LDS address are the DS-usable LDS address).
// The async-load builtin takes (v4i AS1*, v4i AS3*, imm offset, imm cpol).
#define AS1_V4I(p) ((__attribute__((address_space(1))) v4i*)(uintptr_t)(p))
#define AS3_V4I(p) ((__attribute__((address_space(3))) v4i*)(uint32_t)(uintptr_t)(p))

__device__ __forceinline__ uint32_t rotl32(uint32_t x, int n) {
  return (x << n) | (x >> (32 - n));
}

// JAX threefry2x32 (20 rounds), key = (k0,k1), counter = (x0,x1).
__device__ __forceinline__ void threefry2x32(uint32_t k0, uint32_t k1,
                                             uint32_t x0, uint32_t x1,
                                             uint32_t& o0, uint32_t& o1) {
  const uint32_t ks2 = k0 ^ k1 ^ 0x1BD11BDAu;
  x0 += k0; x1 += k1;
#define TF_R4(a,b,c,d)                                  \
  x0 += x1; x1 = rotl32(x1, a); x1 ^= x0;               \
  x0 += x1; x1 = rotl32(x1, b); x1 ^= x0;               \
  x0 += x1; x1 = rotl32(x1, c); x1 ^= x0;               \
  x0 += x1; x1 = rotl32(x1, d); x1 ^= x0;
  TF_R4(13,15,26,6);  x0 += k1;  x1 += ks2 + 1u;
  TF_R4(17,29,16,24); x0 += ks2; x1 += k0 + 2u;
  TF_R4(13,15,26,6);  x0 += k0;  x1 += k1 + 3u;
  TF_R4(17,29,16,24); x0 += k1;  x1 += ks2 + 4u;
  TF_R4(13,15,26,6);  x0 += ks2; x1 += k0 + 5u;
#undef TF_R4
  o0 = x0; o1 = x1;
}

__global__ __launch_bounds__(128)
void probmaxpool_kernel(const float* __restrict__ H, float* __restrict__ out) {
  // LDS: Hexp strip (64ch x 2rows x 128cols), transposed block-sums St[j][c]
  // (stride 66: bank-conflict-free, keeps 8B alignment), denom_base Db[c][j],
  // pooled Pool[j]. Total 99072 B (< 320 KB; 3 WGs/WGP).
  __shared__ __align__(16) float Hs[64 * 256];   // [c][r(0..1)][col] floats
  __shared__ __align__(16) float St[64 * 66];    // St[j*66 + c] = blocksum S[c][j]
  __shared__ __align__(16) float Db[64 * 64];    // Db[c*64 + j] = exp(-max)
  __shared__ __align__(16) float Pool[64];       // cross-channel block sum per j

  const int blk = blockIdx.x;          // 0..1023
  const int b = blk >> 6;              // batch
  const int i = blk & 63;              // block-row (rows 2i, 2i+1)
  const int t = threadIdx.x;           // 0..127

  // ---- Phase 0: strip H[b, :, 2i:2i+2, :] -> LDS (64 KB) -----------------
  // Per channel c the strip chunk is 1024 contiguous bytes at
  // H + (b*64 + c)*16384 + 2i*128 (floats). LDS is packed identically.
  const float* gbase = H + (size_t)b * (64u * 16384u) + (size_t)(2 * i) * 128u;
  for (int q = 0; q < 32; ++q) {
    const int t16 = t + 128 * q;          // 0..4095 : 16-byte unit index
    const int c   = t16 >> 6;             // 64 x 16B per channel chunk
    const int o   = (t16 & 63) << 2;      // float offset within chunk
    const float* g = gbase + (size_t)c * 16384u + o;
    float* l = &Hs[t16 << 2];
#ifdef USE_ASYNC_LDS
    __builtin_amdgcn_global_load_async_to_lds_b128(AS1_V4I(g), AS3_V4I(l), 0, 0);
#else
    *(float4*)l = *(const float4*)g;
#endif
  }
#ifdef USE_ASYNC_LDS
#if __has_builtin(__builtin_amdgcn_s_wait_asynccnt)
  __builtin_amdgcn_s_wait_asynccnt(0);
#else
  asm volatile("s_wait_asynccnt 0" ::: "memory");
#endif
#endif
  __syncthreads();

  // ---- Phase 1: per (c,j): block max, Hexp (in place), blocksum, exp(-max) -
  for (int q = 0; q < 32; ++q) {
    const int cell = t + 128 * q;        // 0..4095
    const int c = cell >> 6;
    const int j = cell & 63;
    float* hp = &Hs[c * 256 + 2 * j];
    float2 h0 = *(float2*)hp;            // row 2i:   cols 2j, 2j+1
    float2 h1 = *(float2*)(hp + 128);    // row 2i+1: cols 2j, 2j+1
    const float m = fmaxf(fmaxf(h0.x, h0.y), fmaxf(h1.x, h1.y));
    const float e0 = __expf(h0.x - m);
    const float e1 = __expf(h0.y - m);
    const float e2 = __expf(h1.x - m);
    const float e3 = __expf(h1.y - m);
    *(float2*)hp         = make_float2(e0, e1);
    *(float2*)(hp + 128) = make_float2(e2, e3);
    St[j * 66 + c] = (e0 + e1) + (e2 + e3);
    Db[c * 64 + j] = __expf(-m);
  }
  __syncthreads();

  // ---- Phase 2: cross-channel reduction via V_WMMA_F32_16X16X4_F32 --------
  // ones(16x4) x S_chunk(4x16) accumulated over 16 chunks => every D row holds
  // the 64-channel column sums for this wave's 16 block-columns.
  {
    const int lane = t & 31;
    const int w    = t >> 5;                   // wave id 0..3
    const int n    = lane & 15;                // column within tile
    const int koff = (lane < 16) ? 0 : 2;      // A/B K placement (VGPR0:K0/K2, VGPR1:K1/K3)
    const int j    = (w << 4) + n;             // global block-column 0..63
    v2f a; a.x = 1.0f; a.y = 1.0f;             // all-ones A (16x4)
    v8f acc = {0.f, 0.f, 0.f, 0.f, 0.f, 0.f, 0.f, 0.f};
#pragma unroll
    for (int k = 0; k < 16; ++k) {
      const int c0 = (k << 2) + koff;          // channel base for this lane half
      v2f bv = *(v2f*)&St[j * 66 + c0];        // {S[c0][j], S[c0+1][j]}
      acc = __builtin_amdgcn_wmma_f32_16x16x4_f32(
          /*neg_a=*/false, a, /*neg_b=*/false, bv,
          /*c_mod=*/(short)0, acc, /*reuse_a=*/false, /*reuse_b=*/false);
    }
    if (lane < 16) Pool[j] = acc[0];           // D[M=0 or 8, N=j%16] = colsum(j)
  }
  __syncthreads();

  // ---- Phase 3: probs, threefry sampling, states; write to global ---------
  float* Pout = out;
  float* Sout = out + (size_t)16777216u;
  for (int q = 0; q < 32; ++q) {
    const int cell = t + 128 * q;
    const int c = cell >> 6;
    const int j = cell & 63;
    const float* hp = &Hs[c * 256 + 2 * j];
    const float2 e0 = *(const float2*)hp;
    const float2 e1 = *(const float2*)(hp + 128);
    const float denom = Db[c * 64 + j] + Pool[j];
    const float invd = 1.0f / denom;
    const float p0 = e0.x * invd, p1 = e0.y * invd;
    const float p2 = e1.x * invd, p3 = e1.y * invd;

    // cumsum over [p0,p1,p2,p3, 1-sum]
    const float cs0 = p0;
    const float cs1 = cs0 + p1;
    const float cs2 = cs1 + p2;
    const float cs3 = cs2 + p3;
    const float cs4 = cs3 + (1.0f - (((p0 + p1) + p2) + p3));

    // JAX randint(key(42), shape (B,C,nr,nc,1), 0, 10001):
    // hi,lo = threefry((0,42), (p, p+N)), N = 16*64*64*64 = 4194304;
    // val = (hi*(2^32 mod 10001) + lo) mod 10001 in uint32 arithmetic.
    const uint32_t p_lin = (uint32_t)(((b * 64 + c) * 64 + i) * 64 + j);
    uint32_t hi, lo;
    threefry2x32(0u, 42u, p_lin, p_lin + 4194304u, hi, lo);
    const uint32_t ri = (hi * 7843u + lo) % 10001u;
    const float u = (float)ri / 10000.0f;

    float cs[5] = {cs0, cs1, cs2, cs3, cs4};
    int idx = 0;
    float best = 3.402823466e38f;
#pragma unroll
    for (int k = 0; k < 5; ++k) {
      float d = cs[k] - u;
      d = (d < 0.0f) ? 5.0f : d;
      if (d < best) { best = d; idx = k; }   // strict < keeps first occurrence
    }
    const float s0 = (idx == 0) ? 1.0f : 0.0f;
    const float s1 = (idx == 1) ? 1.0f : 0.0f;
    const float s2 = (idx == 2) ? 1.0f : 0.0f;
    const float s3 = (idx == 3) ? 1.0f : 0.0f;

    const size_t go = ((size_t)(b * 64 + c) * 128 + (size_t)(2 * i)) * 128 + (size_t)(2 * j);
    *(float2*)&Pout[go]       = make_float2(p0, p1);
    *(float2*)&Pout[go + 128] = make_float2(p2, p3);
    *(float2*)&Sout[go]       = make_float2(s0, s1);
    *(float2*)&Sout[go + 128] = make_float2(s2, s3);
  }
}

extern "C" void kernel_launch(void* const* d_in, const int* in_sizes, int n_in,
                              void* d_out, int out_size, void* d_ws, size_t ws_size,
                              hipStream_t stream) {
  (void)in_sizes; (void)n_in; (void)d_ws; (void)ws_size; (void)out_size;
  const float* H = (const float*)d_in[0];
  float* out = (float*)d_out;
  // 1024 strips: (b in 0..15) x (block-row i in 0..63); 128 threads = 4 waves.
  probmaxpool_kernel<<<dim3(1024), dim3(128), 0, stream>>>(H, out);
}